// Multi_Head_Attention_71382356459538
// MI455X (gfx1250) — compile-verified
//
#include <hip/hip_runtime.h>

typedef float v2f __attribute__((ext_vector_type(2)));
typedef float v8f __attribute__((ext_vector_type(8)));
typedef int   v4i_vs __attribute__((vector_size(16)));   // matches builtin pointee

#define WMMA_F32(a, b, c) \
    __builtin_amdgcn_wmma_f32_16x16x4_f32(false, (a), false, (b), (short)0, (c), false, false)

static __device__ __forceinline__ v2f load2(const float* p) {
    return *reinterpret_cast<const v2f*>(p);
}

// Async global->LDS copy of 16 bytes (CDNA5 GLOBAL_LOAD_ASYNC_TO_LDS_B128,
// tracked by ASYNCcnt). Builtin if the toolchain declares it, else inline asm.
static __device__ __forceinline__ void async_ld_b128(const float* g, float* l) {
#if __has_builtin(__builtin_amdgcn_global_load_async_to_lds_b128)
    __builtin_amdgcn_global_load_async_to_lds_b128(
        (__attribute__((address_space(1))) v4i_vs*)(void*)g,
        (__attribute__((address_space(3))) v4i_vs*)(void*)l, 0, 0);
#else
    const unsigned loff = (unsigned)(size_t)l;  // low 32 bits = LDS byte offset
    asm volatile("global_load_async_to_lds_b128 %0, %1, off"
                 :: "v"(loff), "v"((unsigned long long)(size_t)g)
                 : "memory");
#endif
}

static __device__ __forceinline__ void wait_async0() {
#if __has_builtin(__builtin_amdgcn_s_wait_asynccnt)
    __builtin_amdgcn_s_wait_asynccnt(0);
#else
    asm volatile("s_wait_asynccnt 0x0" ::: "memory");
#endif
}

// ---------------------------------------------------------------------------
// Kernel 1: fused Q/K/V projection.
// X[16384,128] @ {Wq,Wk,Wv}[128,128] -> Q/K/V stored head-major [B*M, N, 16].
// One wave = one 16x16 output tile, 32 WMMA f32 16x16x4 steps over K=128.
// ---------------------------------------------------------------------------
__global__ __launch_bounds__(256)
void qkv_proj_kernel(const float* __restrict__ X,
                     const float* __restrict__ Wq,
                     const float* __restrict__ Wk,
                     const float* __restrict__ Wv,
                     float* __restrict__ Q,
                     float* __restrict__ K,
                     float* __restrict__ V) {
    const int lane  = threadIdx.x & 31;
    const int gwave = blockIdx.x * 8 + (threadIdx.x >> 5);   // 0..24575
    const int mat   = gwave >> 13;                           // 0:Q 1:K 2:V
    const int rem   = gwave & 8191;
    const int row0  = (rem >> 3) << 4;                       // over B*N = 16384
    const int col0  = (rem & 7) << 4;                        // over D = 128

    const float* W  = (mat == 0) ? Wq : (mat == 1) ? Wk : Wv;
    float*      Out = (mat == 0) ? Q  : (mat == 1) ? K  : V;

    const int n16   = lane & 15;
    const int koff  = (lane >> 4) << 1;   // 0 or 2
    const int half8 = (lane >> 4) << 3;   // 0 or 8

    v8f c = {};
    const float* arow = X + (size_t)(row0 + n16) * 128;
#pragma unroll 4
    for (int s = 0; s < 32; ++s) {
        const int k = (s << 2) + koff;
        v2f a = load2(arow + k);
        v2f b;
        b.x = W[(size_t)k * 128 + col0 + n16];
        b.y = W[(size_t)(k + 1) * 128 + col0 + n16];
        c = WMMA_F32(a, b, c);
    }
    // Scatter store into head-major [ (b*8+head), n, dk ].
#pragma unroll
    for (int r = 0; r < 8; ++r) {
        const int grow = row0 + r + half8;    // 0..16383
        const int bb   = grow >> 10;
        const int nn   = grow & 1023;
        const int cc   = col0 + n16;          // 0..127
        const int head = cc >> 4;
        const int kk   = cc & 15;
        Out[((size_t)(bb * 8 + head) * 1024 + nn) * 16 + kk] = c[r];
    }
}

// ---------------------------------------------------------------------------
// Kernel 2: fused flash attention per (b, head, q-tile of 128 rows).
// K/V 16x16 tiles are staged into LDS with async global->LDS B128 copies,
// double-buffered (tile t+1 copy overlaps tile t compute), synced with
// s_wait_asynccnt + block barrier. Each wave owns 16 query rows:
// scores via 4x f32 WMMA, online softmax with 16-lane shfl_xor reductions,
// P tile C-layout -> A-layout via per-wave LDS (s_wait_dscnt ordered),
// O accumulated via 4x f32 WMMA. Output -> H[16384, head*16+dk].
// ---------------------------------------------------------------------------
__global__ __launch_bounds__(256)
void attn_kernel(const float* __restrict__ Q,
                 const float* __restrict__ K,
                 const float* __restrict__ V,
                 const float* __restrict__ Mask,
                 float* __restrict__ H) {
    __shared__ float kbuf[2][16][18];    // padded row stride (bank spread)
    __shared__ float vbuf[2][16][18];
    __shared__ float pbuf[8][16][17];    // per-wave P tile

    const int t     = threadIdx.x;
    const int lane  = t & 31;
    const int w     = t >> 5;
    const int bm    = blockIdx.x >> 3;          // b*8 + head
    const int qt    = blockIdx.x & 7;
    const int b     = bm >> 3;
    const int head  = bm & 7;
    const int q0    = qt * 128 + w * 16;        // wave's query rows [q0, q0+16)

    const float* Qb = Q + (size_t)bm * 1024 * 16;
    const float* Kb = K + (size_t)bm * 1024 * 16;
    const float* Vb = V + (size_t)bm * 1024 * 16;

    const int n16   = lane & 15;
    const int koff  = (lane >> 4) << 1;   // 0 or 2
    const int half8 = (lane >> 4) << 3;   // 0 or 8

    // Staging role: threads 0..63 copy K tile, 64..127 copy V tile.
    // Tile is 1KB contiguous (head-major layout): 64 lanes x 16B.
    const int sWhich = (t >> 6) & 1;      // 0 = K, 1 = V
    const int sIdx   = t & 63;            // 16B chunk index
    const int sRow   = sIdx >> 2;
    const int sCol   = (sIdx & 3) << 2;
    const float* sSrcBase = sWhich ? Vb : Kb;

    // Preload Q A-fragments for all 4 k-steps (d_k = 16).
    v2f qa[4];
#pragma unroll
    for (int s = 0; s < 4; ++s)
        qa[s] = load2(Qb + (size_t)(q0 + n16) * 16 + s * 4 + koff);

    float mrow[8], lrow[8], alpha[8];
    v8f o = {};
#pragma unroll
    for (int r = 0; r < 8; ++r) { mrow[r] = -3.0e38f; lrow[r] = 0.0f; }

    // Prologue: stage tile 0 into buffer 0.
    if (t < 128) {
        float* dst = sWhich ? &vbuf[0][sRow][sCol] : &kbuf[0][sRow][sCol];
        async_ld_b128(sSrcBase + (size_t)sIdx * 4, dst);
    }
    wait_async0();
    __syncthreads();

    int p = 0;
    for (int ti = 0; ti < 64; ++ti) {
        const int kv = ti << 4;
        // Kick off async copy of the next K/V tile into the other buffer.
        if (ti + 1 < 64 && t < 128) {
            float* dst = sWhich ? &vbuf[p ^ 1][sRow][sCol]
                                : &kbuf[p ^ 1][sRow][sCol];
            async_ld_b128(sSrcBase + (size_t)(kv + 16) * 16 + (size_t)sIdx * 4,
                          dst);
        }
        // S = Q @ K^T : B element (k, j) = kbuf[p][j][k] -> per-lane float2.
        v8f s = {};
#pragma unroll
        for (int st = 0; st < 4; ++st) {
            v2f kb = *reinterpret_cast<const v2f*>(&kbuf[p][n16][st * 4 + koff]);
            s = WMMA_F32(qa[st], kb, s);
        }
        // Scale by 1/sqrt(16) and apply mask: s = s*0.25 + (1-mask)*(-1e10).
        const size_t mbase = ((size_t)b * 1024 + (q0 + half8)) * 1024 + kv + n16;
        if (kv + 16 < 1024)
            __builtin_prefetch(Mask + mbase + 16, 0, 1);
#pragma unroll
        for (int r = 0; r < 8; ++r) {
            const float mv = Mask[mbase + (size_t)r * 1024];
            s[r] = s[r] * 0.25f + (1.0f - mv) * (-1.0e10f);
        }
        // Online softmax: each score row lives in 16 lanes of a wave half.
#pragma unroll
        for (int r = 0; r < 8; ++r) {
            float v = s[r];
            v = fmaxf(v, __shfl_xor(v, 1));
            v = fmaxf(v, __shfl_xor(v, 2));
            v = fmaxf(v, __shfl_xor(v, 4));
            v = fmaxf(v, __shfl_xor(v, 8));
            const float mnew = fmaxf(mrow[r], v);
            float pv = __expf(s[r] - mnew);
            s[r] = pv;
            float rs = pv;
            rs += __shfl_xor(rs, 1);
            rs += __shfl_xor(rs, 2);
            rs += __shfl_xor(rs, 4);
            rs += __shfl_xor(rs, 8);
            alpha[r] = __expf(mrow[r] - mnew);
            lrow[r]  = lrow[r] * alpha[r] + rs;
            mrow[r]  = mnew;
        }
        // P tile: C-layout -> LDS -> A-layout fragments (same-wave DS order).
#pragma unroll
        for (int r = 0; r < 8; ++r) {
            pbuf[w][r + half8][n16] = s[r];
            o[r] *= alpha[r];
        }
        asm volatile("s_wait_dscnt 0x0" ::: "memory");
        // O += P @ V  (V read from the staged LDS tile).
#pragma unroll
        for (int st = 0; st < 4; ++st) {
            const int kk = st * 4 + koff;
            v2f pa;
            pa.x = pbuf[w][n16][kk];
            pa.y = pbuf[w][n16][kk + 1];
            v2f vb;
            vb.x = vbuf[p][kk][n16];
            vb.y = vbuf[p][kk + 1][n16];
            o = WMMA_F32(pa, vb, o);
        }
        // Next tile's async copy must have landed before we read it; all
        // waves must be done reading buffer p before it is restaged.
        wait_async0();
        __syncthreads();
        p ^= 1;
    }
    // Normalize and store H[(b*1024+q)*128 + head*16 + dim].
#pragma unroll
    for (int r = 0; r < 8; ++r) {
        const int qrow = q0 + r + half8;
        H[((size_t)(b * 1024 + qrow)) * 128 + head * 16 + n16] = o[r] / lrow[r];
    }
}

// ---------------------------------------------------------------------------
// Kernel 3: output projection. w_o[8,16,128] flattens to Wo[128,128], so
// feature = H[16384,128] @ Wo. One wave per 16x16 tile; 8192 waves.
// ---------------------------------------------------------------------------
__global__ __launch_bounds__(256)
void out_proj_kernel(const float* __restrict__ Hm,
                     const float* __restrict__ Wo,
                     float* __restrict__ Out) {
    const int lane  = threadIdx.x & 31;
    const int gwave = blockIdx.x * 8 + (threadIdx.x >> 5);   // 0..8191
    const int row0  = (gwave >> 3) << 4;
    const int col0  = (gwave & 7) << 4;

    const int n16   = lane & 15;
    const int koff  = (lane >> 4) << 1;
    const int half8 = (lane >> 4) << 3;

    v8f c = {};
    const float* arow = Hm + (size_t)(row0 + n16) * 128;
#pragma unroll 4
    for (int s = 0; s < 32; ++s) {
        const int k = (s << 2) + koff;
        v2f a = load2(arow + k);
        v2f b;
        b.x = Wo[(size_t)k * 128 + col0 + n16];
        b.y = Wo[(size_t)(k + 1) * 128 + col0 + n16];
        c = WMMA_F32(a, b, c);
    }
#pragma unroll
    for (int r = 0; r < 8; ++r)
        Out[(size_t)(row0 + r + half8) * 128 + col0 + n16] = c[r];
}

// ---------------------------------------------------------------------------
extern "C" void kernel_launch(void* const* d_in, const int* in_sizes, int n_in,
                              void* d_out, int out_size, void* d_ws, size_t ws_size,
                              hipStream_t stream) {
    const float* ref  = (const float*)d_in[0];   // [16,1024,128]
    const float* mask = (const float*)d_in[1];   // [16,1,1024,1024]
    const float* wq   = (const float*)d_in[2];   // [128,128]
    const float* wk   = (const float*)d_in[3];   // [128,128]
    const float* wv   = (const float*)d_in[4];   // [128,128]
    const float* wo   = (const float*)d_in[5];   // [8,16,128] == [128,128]
    float*       out  = (float*)d_out;           // [16,1024,128]
    float*       ws   = (float*)d_ws;

    const size_t perMat = (size_t)128 * 1024 * 16;   // 2M floats per Q/K/V
    float* Qw = ws;
    float* Kw = ws + perMat;
    float* Vw = ws + 2 * perMat;
    float* Hw = ws + 3 * perMat;                     // 2M floats

    qkv_proj_kernel<<<3072, 256, 0, stream>>>(ref, wq, wk, wv, Qw, Kw, Vw);
    attn_kernel<<<1024, 256, 0, stream>>>(Qw, Kw, Vw, mask, Hw);
    out_proj_kernel<<<1024, 256, 0, stream>>>(Hw, wo, out);
}